// ScaledDotProductAttention_31215822307973
// MI455X (gfx1250) — compile-verified
//
#include <hip/hip_runtime.h>
#include <cstdint>

// CDNA5 / gfx1250: wave32, WMMA 16x16x4 f32 (fp32 A/B/C/D).
typedef float v2f __attribute__((ext_vector_type(2)));
typedef float v8f __attribute__((ext_vector_type(8)));

#define BATCH 1024
#define HEADS 16
#define SEQ   8
#define DK    32
#define HEADS_TOTAL (BATCH * HEADS)      // 16384
#define WAVES_PER_BLOCK 8
#define INV_SQRT_DK 0.17677669529663687f // 1/sqrt(32)

// Per-wave LDS layout (floats). Q/K padded to 16 rows so WMMA M/N rows 8..15
// read zeros; W (weights) padded to 16 rows for the context-stage A operand.
#define QS_OFF 0      // 16x32 = 512 floats (rows 8..15 zero; reused for ctx staging)
#define KS_OFF 512    // 16x32 = 512 floats (rows 8..15 zero)
#define VS_OFF 1024   // 8x32  = 256 floats
#define WS_OFF 1280   // 16x8  = 128 floats (rows 8..15 zero)
#define WAVE_LDS 1408 // floats per wave

__global__ __launch_bounds__(256)
void sdpa_wmma_f32_kernel(const float* __restrict__ Qg_,
                          const float* __restrict__ Kg_,
                          const float* __restrict__ Vg_,
                          const uint8_t* __restrict__ mask, // jnp bool = 1 byte/elt
                          const float* __restrict__ hyper,
                          float* __restrict__ ctx_out,
                          float* __restrict__ w_out)
{
    __shared__ float smem[WAVE_LDS * WAVES_PER_BLOCK]; // 44 KB / block
    const int wave = threadIdx.x >> 5;
    const int lane = threadIdx.x & 31;
    const int bh   = blockIdx.x * WAVES_PER_BLOCK + wave; // one head per wave
    float* S = smem + wave * WAVE_LDS;

    const int hi16 = lane >> 4;  // 0: lanes 0-15, 1: lanes 16-31
    const int mrow = lane & 15;  // M (or N) index within 16x16 tile

    // ---- Phase 1: stage Q,K,V (8x32 f32 each) into LDS; zero padding rows ----
    {
        const float4* Qg = (const float4*)(Qg_ + (size_t)bh * (SEQ * DK));
        const float4* Kg = (const float4*)(Kg_ + (size_t)bh * (SEQ * DK));
        const float4* Vg = (const float4*)(Vg_ + (size_t)bh * (SEQ * DK));
        float4* Qs4 = (float4*)(S + QS_OFF);
        float4* Ks4 = (float4*)(S + KS_OFF);
        float4* Vs4 = (float4*)(S + VS_OFF);
        Qs4[lane]      = Qg[lane];
        Qs4[lane + 32] = Qg[lane + 32];
        Ks4[lane]      = Kg[lane];
        Ks4[lane + 32] = Kg[lane + 32];
        Vs4[lane]      = Vg[lane];
        Vs4[lane + 32] = Vg[lane + 32];
        float4 z; z.x = z.y = z.z = z.w = 0.0f;
        Qs4[lane + 64] = z;  Qs4[lane + 96] = z;   // Q rows 8..15 = 0
        Ks4[lane + 64] = z;  Ks4[lane + 96] = z;   // K rows 8..15 = 0
        if (lane < 16) ((float4*)(S + WS_OFF + 64))[lane] = z; // W rows 8..15 = 0
    }
    __syncthreads();

    // ---- Phase 2: scores S = Q * K^T via 8 chained v_wmma_f32_16x16x4_f32 ----
    // A(16x4): lane m=mrow holds K = kk*4 + 2*hi16 + {0,1}.
    // B(4x16): lane n=mrow holds the same K slices of the K-tile -> identical addressing.
    v8f acc = {};
#pragma unroll
    for (int kk = 0; kk < 8; ++kk) {
        const int off = kk * 4 + hi16 * 2;
        v2f a = *(const v2f*)(S + QS_OFF + mrow * DK + off);
        v2f b = *(const v2f*)(S + KS_OFF + mrow * DK + off);
        acc = __builtin_amdgcn_wmma_f32_16x16x4_f32(false, a, false, b,
                                                    (short)0, acc, false, false);
    }

    // ---- Phase 3: spill valid 8x8 score tile to LDS (lane = N/k, VGPR = M/q) ----
    if (lane < SEQ) {
#pragma unroll
        for (int g = 0; g < 8; ++g) S[WS_OFF + g * SEQ + lane] = acc[g];
    }
    __syncthreads();

    // ---- Phase 4: scale + mask + softmax, one q-row per lane (lanes 0-7) ----
    if (lane < SEQ) {
        const int q = lane;
        const uint32_t* mp = (const uint32_t*)(mask + (size_t)bh * 64 + q * 8);
        const uint32_t m0 = mp[0], m1 = mp[1];
        float s[8];
        float mx = -3.0e38f;
#pragma unroll
        for (int k = 0; k < 8; ++k) {
            const uint32_t mb = (k < 4) ? ((m0 >> (8 * k)) & 0xFFu)
                                        : ((m1 >> (8 * (k - 4))) & 0xFFu);
            const float raw = S[WS_OFF + q * SEQ + k] * INV_SQRT_DK;
            const float v = mb ? -1.0e9f : raw;
            s[k] = v;
            mx = fmaxf(mx, v);
        }
        float sum = 0.0f;
#pragma unroll
        for (int k = 0; k < 8; ++k) { const float e = __expf(s[k] - mx); s[k] = e; sum += e; }
        const float inv = 1.0f / sum;
#pragma unroll
        for (int k = 0; k < 8; ++k) S[WS_OFF + q * SEQ + k] = s[k] * inv;
    }
    __syncthreads();

    // ---- Phase 5: store weights output (8x8 = 16 float4, coalesced) ----
    if (lane < 16) {
        float4* wo = (float4*)(w_out + (size_t)bh * 64);
        wo[lane] = ((const float4*)(S + WS_OFF))[lane];
    }

    // ---- Phase 6: context = W(8x8) * V(8x32): 2 N-halves x 2 K-chunks WMMAs ----
    v8f clo = {}, chi = {};
#pragma unroll
    for (int kk = 0; kk < 2; ++kk) {
        const int off = kk * 4 + hi16 * 2; // K rows handled by this lane half
        v2f a = *(const v2f*)(S + WS_OFF + mrow * SEQ + off);
        v2f blo, bhi;
        blo.x = S[VS_OFF + off * DK + mrow];
        blo.y = S[VS_OFF + (off + 1) * DK + mrow];
        bhi.x = S[VS_OFF + off * DK + 16 + mrow];
        bhi.y = S[VS_OFF + (off + 1) * DK + 16 + mrow];
        clo = __builtin_amdgcn_wmma_f32_16x16x4_f32(false, a, false, blo,
                                                    (short)0, clo, false, false);
        chi = __builtin_amdgcn_wmma_f32_16x16x4_f32(false, a, false, bhi,
                                                    (short)0, chi, false, false);
    }

    // ---- Phase 7: stage 8x32 context into LDS (reuse Q region) ----
    if (lane < 16) {
#pragma unroll
        for (int g = 0; g < 8; ++g) {
            S[QS_OFF + g * DK + mrow]      = clo[g];
            S[QS_OFF + g * DK + 16 + mrow] = chi[g];
        }
    }
    __syncthreads();

    // ---- Phase 8: fused hyper scale + coalesced float4 store ----
    {
        const float4* h4 = (const float4*)(hyper + (size_t)bh * 256);
        float4*       o4 = (float4*)(ctx_out + (size_t)bh * 256);
        const float4* c4 = (const float4*)(S + QS_OFF);
#pragma unroll
        for (int t = 0; t < 2; ++t) {
            const int i = lane + t * 32;
            const float4 c = c4[i];
            const float4 h = h4[i];
            float4 r;
            r.x = c.x * h.x; r.y = c.y * h.y; r.z = c.z * h.z; r.w = c.w * h.w;
            o4[i] = r;
        }
    }
}

extern "C" void kernel_launch(void* const* d_in, const int* in_sizes, int n_in,
                              void* d_out, int out_size, void* d_ws, size_t ws_size,
                              hipStream_t stream) {
    (void)in_sizes; (void)n_in; (void)out_size; (void)d_ws; (void)ws_size;
    const float*   Q     = (const float*)d_in[0];
    const float*   K     = (const float*)d_in[1];
    const float*   V     = (const float*)d_in[2];
    const uint8_t* mask  = (const uint8_t*)d_in[3]; // bool array, 1 byte/elt
    const float*   hyper = (const float*)d_in[4];

    float* ctx = (float*)d_out;                                    // [B,H,S,DK]
    float* wts = (float*)d_out + (size_t)BATCH * HEADS * SEQ * DK; // [B,H,S,S]

    dim3 grid(HEADS_TOTAL / WAVES_PER_BLOCK); // 2048 blocks
    dim3 block(256);                          // 8 wave32 = 8 heads per block
    sdpa_wmma_f32_kernel<<<grid, block, 0, stream>>>(Q, K, V, mask, hyper, ctx, wts);
}